// EvidenceRetriever_43009802502433
// MI455X (gfx1250) — compile-verified
//
#include <hip/hip_runtime.h>
#include <math.h>

#define ET 16384
#define ES 32768
#define FD 512
#define TOPK 8
#define W_IMP 0.1f

#define WAVES 4
#define TGT_PER_WG (WAVES * 16)   // 64 targets per workgroup
#define NBLK (ES / 16)            // 2048 source blocks of 16
#define LDSW 516                  // effective row stride (dwords) after TDM padding

typedef __attribute__((ext_vector_type(2))) float v2f;
typedef __attribute__((ext_vector_type(8))) float v8f;
typedef __attribute__((ext_vector_type(4))) unsigned int u32x4;
typedef __attribute__((ext_vector_type(8))) unsigned int u32x8;

// ---------------- Pass 1: inverse norms + source meta ----------------
__global__ void __launch_bounds__(256)
er_norms(const float* __restrict__ tgt, const float* __restrict__ src,
         const float* __restrict__ imp, float* __restrict__ ws) {
    const int wave = threadIdx.x >> 5;
    const int lane = threadIdx.x & 31;
    const int row  = blockIdx.x * 8 + wave;   // one wave32 per row
    const float* p = (row < ET) ? (tgt + (size_t)row * FD)
                                : (src + (size_t)(row - ET) * FD);
    float ss = 0.f;
#pragma unroll
    for (int c = 0; c < 4; ++c) {
        const float4 v = ((const float4*)p)[lane + c * 32];
        ss += v.x * v.x + v.y * v.y + v.z * v.z + v.w * v.w;
    }
#pragma unroll
    for (int off = 16; off >= 1; off >>= 1) ss += __shfl_xor(ss, off, 32);
    const float inv = 1.0f / fmaxf(sqrtf(ss), 1e-12f);
    if (lane == 0) {
        if (row < ET) {
            ws[row] = inv;
        } else {
            const int s = row - ET;
            ws[ET + 2 * s]     = inv;
            ws[ET + 2 * s + 1] = W_IMP * imp[s];
        }
    }
}

// ---- TDM: DMA a (rows x 512) f32 row-major tile global->LDS.
// Pad descriptor: pad_enable, interval=256 dwords (code 7), amount=2 dwords (code 1)
// => LDS layout: element (r,k) at dword  r*516 + k + 2*(k>=256)  (bank-conflict-free).
__device__ __forceinline__ void tdm_load_2d(unsigned lds_off, const void* gaddr,
                                            unsigned rows) {
    const unsigned long long ga = (unsigned long long)gaddr;
    u32x4 g0;
    g0[0] = 1u;                                   // count=1 (valid descriptor)
    g0[1] = lds_off;                              // lds_addr (bytes)
    g0[2] = (unsigned)ga;                         // global_addr[31:0]
    g0[3] = (unsigned)(ga >> 32) | (2u << 30);    // global_addr[56:32] | type=2
    u32x8 g1;
    g1[0] = (2u << 16)      // data_size = 4B
          | (1u << 20)      // pad_enable
          | (7u << 22)      // pad_interval: 256 dwords
          | (1u << 25);     // pad_amount: 2 dwords
    g1[1] = (FD & 0xFFFFu) << 16;   // tensor_dim0 = 512 (bits 48..63)
    g1[2] = (rows << 16);           // tensor_dim0 hi | tensor_dim1 = rows
    g1[3] = ((unsigned)FD << 16);   // tensor_dim1 hi | tile_dim0 = 512
    g1[4] = rows;                   // tile_dim1 = rows, tile_dim2 = 0
    g1[5] = (unsigned)FD;           // tensor_dim0_stride = 512 (lo 32)
    g1[6] = 0u;                     // stride0 hi | tensor_dim1_stride lo
    g1[7] = 0u;
    asm volatile("tensor_load_to_lds %0, %1" :: "s"(g0), "s"(g1) : "memory");
}

// ---------------- Pass 2: fused fp32-WMMA GEMM + top-8 + softmax ----------------
// M = sources (16/block), N = targets (16/wave), K = 512.
// D layout (16x16 f32): VGPR r, lanes 0-15 -> (M=r, N=lane); lanes 16-31 -> (M=r+8).
// => lane-local running top-8 per target.
__global__ void __launch_bounds__(128)
er_topk(const float* __restrict__ tgt, const float* __restrict__ src,
        const float* __restrict__ ws, float* __restrict__ out) {
    __shared__ __align__(16) float ldsT[TGT_PER_WG * LDSW];  // 132096 B
    __shared__ __align__(16) float ldsS[2][16 * LDSW];       // 2 x 33024 B

    const int tid  = threadIdx.x;
    const int lane = tid & 31;
    const int wave = tid >> 5;
    const int half = lane >> 4;
    const int l16  = lane & 15;
    const int t0   = blockIdx.x * TGT_PER_WG;

    // Prologue DMAs: whole target block + first source tile (wave 0 issues once)
    if (wave == 0) {
        tdm_load_2d((unsigned)(size_t)&ldsT[0], tgt + (size_t)t0 * FD, TGT_PER_WG);
        tdm_load_2d((unsigned)(size_t)&ldsS[0][0], src, 16);
    }

    const float  invT  = ws[t0 + wave * 16 + l16];
    const float2* meta = (const float2*)(ws + ET);

    float vals[TOPK];
    int   idxs[TOPK];
#pragma unroll
    for (int i = 0; i < TOPK; ++i) { vals[i] = -1e30f; idxs[i] = 0; }

    const int fragOff = l16 * LDSW + 2 * half;                // A (source) fragment base
    const float* bBase = &ldsT[(wave * 16 + l16) * LDSW + 2 * half];

    int cur = 0;
    for (int blk = 0; blk < NBLK; ++blk) {
        const int s0 = blk * 16;
        if (wave == 0) __builtin_amdgcn_s_wait_tensorcnt(0);  // tile `blk` landed
        __syncthreads();  // everyone sees it; everyone done reading buf cur^1

        // Prefetch next tile into the other buffer (overlaps with compute below)
        if (wave == 0 && blk + 1 < NBLK)
            tdm_load_2d((unsigned)(size_t)&ldsS[cur ^ 1][0],
                        src + (size_t)(blk + 1) * 16 * FD, 16);

        // Prefetch per-source (invS, bias) — latency hidden by the WMMA loop
        float2 m[8];
#pragma unroll
        for (int r = 0; r < 8; ++r) m[r] = meta[s0 + 8 * half + r];

        const float* aBase = &ldsS[cur][fragOff];
        v8f acc0 = {0.f, 0.f, 0.f, 0.f, 0.f, 0.f, 0.f, 0.f};
        v8f acc1 = {0.f, 0.f, 0.f, 0.f, 0.f, 0.f, 0.f, 0.f};
        // Two interleaved K-half chains (independent accumulators).
        // Second half sits +2 dwords further due to the TDM mid-row pad.
#pragma unroll 8
        for (int k = 0; k < 256; k += 4) {
            const v2f a0 = *(const v2f*)(aBase + k);
            const v2f b0 = *(const v2f*)(bBase + k);
            acc0 = __builtin_amdgcn_wmma_f32_16x16x4_f32(
                       false, a0, false, b0, (short)0, acc0, false, false);
            const v2f a1 = *(const v2f*)(aBase + k + 258);
            const v2f b1 = *(const v2f*)(bBase + k + 258);
            acc1 = __builtin_amdgcn_wmma_f32_16x16x4_f32(
                       false, a1, false, b1, (short)0, acc1, false, false);
        }

        // Epilogue: combine halves, scale by norms, add importance bias
        float sc[8];
#pragma unroll
        for (int r = 0; r < 8; ++r) {
            const float d = acc0[r] + acc1[r];
            sc[r] = fmaf(d * m[r].x, invT, m[r].y);
        }
        float mx = sc[0];
#pragma unroll
        for (int r = 1; r < 8; ++r) mx = fmaxf(mx, sc[r]);

        if (__any(mx > vals[TOPK - 1])) {
#pragma unroll
            for (int r = 0; r < 8; ++r) {
                const float c  = sc[r];
                const int   ci = s0 + 8 * half + r;
                if (c > vals[TOPK - 1]) {
                    vals[TOPK - 1] = c; idxs[TOPK - 1] = ci;
#pragma unroll
                    for (int j = TOPK - 1; j >= 1; --j) {
                        const bool sw = vals[j] > vals[j - 1];
                        const float tv = vals[j - 1]; const int ti = idxs[j - 1];
                        vals[j - 1] = sw ? vals[j] : vals[j - 1];
                        idxs[j - 1] = sw ? idxs[j] : idxs[j - 1];
                        vals[j] = sw ? tv : vals[j];
                        idxs[j] = sw ? ti : idxs[j];
                    }
                }
            }
        }
        cur ^= 1;
    }

    // Merge lane L with lane L^16 (same target, disjoint source halves)
    float bv[TOPK]; int bi[TOPK];
#pragma unroll
    for (int i = 0; i < TOPK; ++i) {
        bv[i] = __int_as_float(__builtin_amdgcn_ds_swizzle(__float_as_int(vals[i]), 0x401F));
        bi[i] = __builtin_amdgcn_ds_swizzle(idxs[i], 0x401F);   // SWAPX16
    }
    float mv[TOPK]; int mi[TOPK];
#pragma unroll
    for (int i = 0; i < TOPK; ++i) {
        const bool t = vals[i] >= bv[TOPK - 1 - i];
        mv[i] = t ? vals[i] : bv[TOPK - 1 - i];
        mi[i] = t ? idxs[i] : bi[TOPK - 1 - i];
    }
#define CE(i, j)                                                            \
    { const bool s_ = mv[j] > mv[i];                                        \
      const float tv_ = mv[i]; const int ti_ = mi[i];                       \
      mv[i] = s_ ? mv[j] : mv[i]; mi[i] = s_ ? mi[j] : mi[i];               \
      mv[j] = s_ ? tv_ : mv[j];  mi[j] = s_ ? ti_ : mi[j]; }
    CE(0, 4) CE(1, 5) CE(2, 6) CE(3, 7)
    CE(0, 2) CE(1, 3) CE(4, 6) CE(5, 7)
    CE(0, 1) CE(2, 3) CE(4, 5) CE(6, 7)
#undef CE

    float e[TOPK], sum = 0.f;
#pragma unroll
    for (int i = 0; i < TOPK; ++i) { e[i] = __expf(mv[i] - mv[0]); sum += e[i]; }
    const float rinv = 1.0f / sum;

    if (half == 0) {
        const int t = t0 + wave * 16 + l16;
        float* oidx = out;
        float* oscr = out + (size_t)ET * TOPK;
        float* oal  = out + (size_t)2 * ET * TOPK;
#pragma unroll
        for (int i = 0; i < TOPK; ++i) {
            oidx[(size_t)t * TOPK + i] = (float)mi[i];
            oscr[(size_t)t * TOPK + i] = mv[i];
            oal[(size_t)t * TOPK + i]  = e[i] * rinv;
        }
    }
}

extern "C" void kernel_launch(void* const* d_in, const int* in_sizes, int n_in,
                              void* d_out, int out_size, void* d_ws, size_t ws_size,
                              hipStream_t stream) {
    (void)in_sizes; (void)n_in; (void)out_size; (void)ws_size;
    const float* tgt = (const float*)d_in[0];
    const float* src = (const float*)d_in[1];
    const float* imp = (const float*)d_in[2];
    float* ws  = (float*)d_ws;   // needs (ET + 2*ES)*4 = 320 KB
    float* out = (float*)d_out;

    er_norms<<<(ET + ES) / 8, 256, 0, stream>>>(tgt, src, imp, ws);
    er_topk<<<ET / TGT_PER_WG, 128, 0, stream>>>(tgt, src, ws, out);
}